// AttractorPooling_29326036697494
// MI455X (gfx1250) — compile-verified
//
#include <hip/hip_runtime.h>
#include <hip/hip_bf16.h>
#include <math.h>

typedef __attribute__((ext_vector_type(2))) float v2f;
typedef __attribute__((ext_vector_type(8))) float v8f;

#define T_N 1000
#define T_PAD 1024
#define NTILE 63          // ceil(1008/16): rows 0..1007, cols padded with 1e4
#define B_N 32
#define NEPS 4
#define OUTD 256
#define FEATD 29

// ---------------------------------------------------------------------------
// Kernel 1: per-row threshold counts c_r(eps) via WMMA f32 16x16x4 Gram tiles
// ---------------------------------------------------------------------------
__global__ __launch_bounds__(1024) void k_count(const float* __restrict__ x,
                                                unsigned* __restrict__ counts) {
  __shared__ float xs0[T_PAD], xs1[T_PAD], xs2[T_PAD], sq[T_PAD];
  const int b = blockIdx.x;
  const int tid = threadIdx.x;
  {
    float a0 = 1e4f, a1 = 1e4f, a2 = 1e4f;   // far-away padding: never < eps
    if (tid < T_N) {
      const float* p = x + (size_t)(b * T_N + tid) * 3;
      a0 = p[0]; a1 = p[1]; a2 = p[2];
    }
    xs0[tid] = a0; xs1[tid] = a1; xs2[tid] = a2;
    sq[tid] = a0 * a0 + a1 * a1 + a2 * a2;
  }
  __syncthreads();

  const int wave = tid >> 5;
  const int lane = tid & 31;
  const int half = lane >> 4;     // wave32: lanes 0-15 vs 16-31
  const int l = lane & 15;
  const float eps2[NEPS] = {1e-4f, 1e-2f, 0.25f, 1.0f};

  for (int it = wave; it < NTILE; it += 32) {
    const int i0 = it * 16;
    // A-matrix 16x4 f32: lanes0-15 hold K=0 (v0), K=1 (v1); lanes16-31 K=2, K=3(=0 pad)
    v2f A;
    A.x = half ? xs2[i0 + l] : xs0[i0 + l];
    A.y = half ? 0.0f        : xs1[i0 + l];
    float si[8];
#pragma unroll
    for (int v = 0; v < 8; ++v) si[v] = sq[i0 + v + half * 8];
    unsigned cnt[NEPS][8];
#pragma unroll
    for (int e = 0; e < NEPS; ++e)
#pragma unroll
      for (int v = 0; v < 8; ++v) cnt[e][v] = 0u;

    for (int jt = 0; jt < NTILE; ++jt) {
      const int j0 = jt * 16;
      // B-matrix 4x16 f32: N on lanes (mirror of A layout)
      v2f Bv;
      Bv.x = half ? xs2[j0 + l] : xs0[j0 + l];
      Bv.y = half ? 0.0f        : xs1[j0 + l];
      const float sj = sq[j0 + l];
      v8f g = {};
#if __has_builtin(__builtin_amdgcn_wmma_f32_16x16x4_f32)
      g = __builtin_amdgcn_wmma_f32_16x16x4_f32(false, A, false, Bv,
                                                (short)0, g, false, false);
#else
#pragma unroll
      for (int v = 0; v < 8; ++v) {
        const int row = i0 + v + half * 8;
        const float dx0 = xs0[row] - xs0[j0 + l];
        const float dx1 = xs1[row] - xs1[j0 + l];
        const float dx2 = xs2[row] - xs2[j0 + l];
        // store -0.5*d2 + 0.5*(si+sj) into g so the formula below matches
        g[v] = 0.5f * (si[v] + sj - (dx0 * dx0 + dx1 * dx1 + dx2 * dx2));
      }
#endif
#pragma unroll
      for (int v = 0; v < 8; ++v) {
        const float d2 = si[v] + sj - 2.0f * g[v];   // sqrt-free: d<eps <=> d2<eps^2
#pragma unroll
        for (int e = 0; e < NEPS; ++e) cnt[e][v] += (d2 < eps2[e]) ? 1u : 0u;
      }
    }
    // reduce across the 16 lanes of each half-wave (rows split per C/D layout)
#pragma unroll
    for (int e = 0; e < NEPS; ++e)
#pragma unroll
      for (int v = 0; v < 8; ++v) {
        unsigned c = cnt[e][v];
        c += __shfl_xor(c, 1, 32);
        c += __shfl_xor(c, 2, 32);
        c += __shfl_xor(c, 4, 32);
        c += __shfl_xor(c, 8, 32);
        cnt[e][v] = c;
      }
    if (l == 0) {
#pragma unroll
      for (int v = 0; v < 8; ++v) {
        const int row = i0 + v + half * 8;
        if (row < T_N) {
#pragma unroll
          for (int e = 0; e < NEPS; ++e)
            counts[((b * NEPS + e) << 10) + row] = cnt[e][v];
        }
      }
    }
  }
}

// ---------------------------------------------------------------------------
// Kernel 2: per-(b,eps) run-length entropy over diagonals + correlation dim
// ---------------------------------------------------------------------------
__global__ __launch_bounds__(256) void k_entropy(const unsigned* __restrict__ counts,
                                                 float* __restrict__ corr,
                                                 float* __restrict__ ents) {
  __shared__ unsigned c[T_N];
  __shared__ unsigned hist[T_N + 1];
  __shared__ unsigned sTotC;
  __shared__ float sTot, sH;
  const int be = blockIdx.x;        // b*4 + e
  const int e = be & 3;
  const int tid = threadIdx.x;
  if (tid == 0) { sTotC = 0u; sTot = 0.0f; sH = 0.0f; }
  unsigned partial = 0;
  for (int t = tid; t < T_N; t += 256) {
    unsigned v = counts[(be << 10) + t];
    c[t] = v; partial += v;
  }
  for (int i = tid; i <= T_N; i += 256) hist[i] = 0u;
  __syncthreads();
  atomicAdd(&sTotC, partial);

  // 1996 diagonals: k<998 -> o=k-998 (negative); else o=k-997 (positive)
  for (int k = tid; k < 1996; k += 256) {
    unsigned run = 0;
    if (k < 998) {
      const int q = 998 - k;               // |o|
      const int len = T_N - q;
      for (int t = 0; t < len; ++t) {
        const bool v = ((unsigned)t < c[t + q]);
        if (v) { ++run; }
        else { if (run >= 2u) atomicAdd(&hist[run], 1u); run = 0; }
      }
    } else {
      const int o = k - 997;
      const int len = T_N - o;
      for (int t = 0; t < len; ++t) {
        const bool v = ((unsigned)(t + o) < c[t]);
        if (v) { ++run; }
        else { if (run >= 2u) atomicAdd(&hist[run], 1u); run = 0; }
      }
    }
    if (run >= 2u) atomicAdd(&hist[run], 1u);
  }
  __syncthreads();

  float pTot = 0.0f;
  for (int i = tid; i <= T_N; i += 256) pTot += (float)hist[i];
  atomicAdd(&sTot, pTot);
  __syncthreads();
  const float tot = sTot;
  float pH = 0.0f;
  for (int i = tid; i <= T_N; i += 256) {
    if (hist[i] > 0u) {
      const float p = (float)hist[i] / fmaxf(tot, 1.0f);
      pH += p * logf(fmaxf(p, 1e-30f));
    }
  }
  atomicAdd(&sH, pH);
  __syncthreads();
  if (tid == 0) {
    float H = -sH;
    H = fminf(fmaxf(H, 0.0f), 10.0f);
    ents[be] = (tot > 0.0f) ? H : 0.0f;

    const float epsv[NEPS] = {0.01f, 0.1f, 0.5f, 1.0f};
    const float cs = ((float)sTotC - (float)T_N) / (float)(T_N * (T_N - 1));
    float cd = logf(fmaxf(cs, 1e-30f)) / logf(epsv[e]);
    // NaN-preserving clip (jnp.clip(NaN)=NaN -> nan_to_num -> 0 downstream)
    cd = (cd < -10.0f) ? -10.0f : ((cd > 10.0f) ? 10.0f : cd);
    corr[be] = (cs > 1e-10f) ? cd : 0.0f;
  }
}

// ---------------------------------------------------------------------------
// Kernel 3: per-batch per-dim stats (mean/std/max/min/median/kurt/skew)
// ---------------------------------------------------------------------------
__global__ __launch_bounds__(1024) void k_stats(const float* __restrict__ x,
                                                float* __restrict__ stats) {
  __shared__ float red[1024];
  const int b = blockIdx.x, tid = threadIdx.x;
  for (int d = 0; d < 3; ++d) {
    const bool valid = tid < T_N;
    const float v = valid ? x[(size_t)(b * T_N + tid) * 3 + d] : 0.0f;

    red[tid] = valid ? v : 0.0f; __syncthreads();
    for (int s = 512; s > 0; s >>= 1) { if (tid < s) red[tid] += red[tid + s]; __syncthreads(); }
    const float mean = red[0] / (float)T_N; __syncthreads();

    const float cc = v - mean;
    red[tid] = valid ? cc * cc : 0.0f; __syncthreads();
    for (int s = 512; s > 0; s >>= 1) { if (tid < s) red[tid] += red[tid + s]; __syncthreads(); }
    const float m2 = red[0] / (float)T_N; __syncthreads();

    red[tid] = valid ? cc * cc * cc : 0.0f; __syncthreads();
    for (int s = 512; s > 0; s >>= 1) { if (tid < s) red[tid] += red[tid + s]; __syncthreads(); }
    const float m3 = red[0] / (float)T_N; __syncthreads();

    red[tid] = valid ? cc * cc * cc * cc : 0.0f; __syncthreads();
    for (int s = 512; s > 0; s >>= 1) { if (tid < s) red[tid] += red[tid + s]; __syncthreads(); }
    const float m4 = red[0] / (float)T_N; __syncthreads();

    red[tid] = valid ? v : -INFINITY; __syncthreads();
    for (int s = 512; s > 0; s >>= 1) { if (tid < s) red[tid] = fmaxf(red[tid], red[tid + s]); __syncthreads(); }
    const float mx = red[0]; __syncthreads();

    red[tid] = valid ? v : INFINITY; __syncthreads();
    for (int s = 512; s > 0; s >>= 1) { if (tid < s) red[tid] = fminf(red[tid], red[tid + s]); __syncthreads(); }
    const float mn = red[0]; __syncthreads();

    // median via bitonic sort of 1024 (+inf padded)
    red[tid] = valid ? v : INFINITY; __syncthreads();
    for (int k = 2; k <= 1024; k <<= 1) {
      for (int j = k >> 1; j > 0; j >>= 1) {
        const int ixj = tid ^ j;
        if (ixj > tid) {
          const float a = red[tid], bb = red[ixj];
          const bool up = ((tid & k) == 0);
          if ((a > bb) == up) { red[tid] = bb; red[ixj] = a; }
        }
        __syncthreads();
      }
    }
    const float med = 0.5f * (red[499] + red[500]); __syncthreads();

    if (tid == 0) {
      float* o = stats + b * 21;
      o[0 * 3 + d] = mean;
      o[1 * 3 + d] = sqrtf(m2);
      o[2 * 3 + d] = mx;
      o[3 * 3 + d] = mn;
      o[4 * 3 + d] = med;
      o[5 * 3 + d] = m4 / fmaxf(m2 * m2, 1e-30f) - 3.0f;
      o[6 * 3 + d] = m3 / fmaxf(powf(m2, 1.5f), 1e-30f);
    }
    __syncthreads();
  }
}

// ---------------------------------------------------------------------------
// Kernel 4: feature assembly (nan_to_num) + 29x256 GEMM + LayerNorm
// ---------------------------------------------------------------------------
__global__ __launch_bounds__(256) void k_head(const float* __restrict__ corr,
                                              const float* __restrict__ ents,
                                              const float* __restrict__ stats,
                                              const float* __restrict__ W,
                                              const float* __restrict__ bias,
                                              const float* __restrict__ gamma,
                                              const float* __restrict__ beta,
                                              float* __restrict__ out) {
  __shared__ float f[FEATD];
  __shared__ float red[OUTD];
  const int b = blockIdx.x, tid = threadIdx.x;
  if (tid < FEATD) {
    float v;
    if (tid < 4)      v = corr[b * 4 + tid];
    else if (tid < 8) v = ents[b * 4 + tid - 4];
    else              v = stats[b * 21 + tid - 8];
    if (isnan(v)) v = 0.0f;
    else if (isinf(v)) v = (v > 0.0f) ? 1e6f : -1e6f;
    f[tid] = v;
  }
  __syncthreads();
  float y = bias[tid];
#pragma unroll
  for (int k = 0; k < FEATD; ++k) y += f[k] * W[k * OUTD + tid];

  red[tid] = y; __syncthreads();
  for (int s = 128; s > 0; s >>= 1) { if (tid < s) red[tid] += red[tid + s]; __syncthreads(); }
  const float mu = red[0] / (float)OUTD; __syncthreads();
  const float c = y - mu;
  red[tid] = c * c; __syncthreads();
  for (int s = 128; s > 0; s >>= 1) { if (tid < s) red[tid] += red[tid + s]; __syncthreads(); }
  const float var = red[0] / (float)OUTD;
  out[b * OUTD + tid] = c * rsqrtf(var + 1e-5f) * gamma[tid] + beta[tid];
}

// ---------------------------------------------------------------------------
extern "C" void kernel_launch(void* const* d_in, const int* in_sizes, int n_in,
                              void* d_out, int out_size, void* d_ws, size_t ws_size,
                              hipStream_t stream) {
  const float* x     = (const float*)d_in[0];   // (32,1000,3)
  const float* W     = (const float*)d_in[1];   // (29,256)
  const float* bias  = (const float*)d_in[2];   // (256)
  const float* gamma = (const float*)d_in[3];   // (256)
  const float* beta  = (const float*)d_in[4];   // (256)
  float* out = (float*)d_out;                   // (32,256)

  char* ws = (char*)d_ws;
  unsigned* counts = (unsigned*)ws;                              // 32*4*1024 u32
  float* corr  = (float*)(ws + (size_t)B_N * NEPS * 1024 * 4);   // 32*4
  float* ents  = corr + B_N * NEPS;                              // 32*4
  float* stats = ents + B_N * NEPS;                              // 32*21

  k_count  <<<B_N, 1024, 0, stream>>>(x, counts);
  k_entropy<<<B_N * NEPS, 256, 0, stream>>>(counts, corr, ents);
  k_stats  <<<B_N, 1024, 0, stream>>>(x, stats);
  k_head   <<<B_N, 256, 0, stream>>>(corr, ents, stats, W, bias, gamma, beta, out);
}